// FocalLoss_56367150792829
// MI455X (gfx1250) — compile-verified
//
#include <hip/hip_runtime.h>
#include <hip/hip_bf16.h>
#include <stdint.h>

#define BB 8
#define AA 100000
#define CC 80
#define MM 32

typedef float f4  __attribute__((ext_vector_type(4)));
typedef float v2f __attribute__((ext_vector_type(2)));
typedef float v8f __attribute__((ext_vector_type(8)));

#if __has_builtin(__builtin_amdgcn_global_load_async_to_lds_b32)
#define ASYNC_LDS 1
#else
#define ASYNC_LDS 0
#endif

__device__ __forceinline__ void wait_async0() {
#if ASYNC_LDS
#if __has_builtin(__builtin_amdgcn_s_wait_asynccnt)
    __builtin_amdgcn_s_wait_asynccnt(0);
#else
    asm volatile("s_wait_asynccnt 0" ::: "memory");
#endif
#endif
}

__device__ __forceinline__ float wave_red_f(float v) {
    #pragma unroll
    for (int m = 16; m > 0; m >>= 1) v += __shfl_xor(v, m, 32);
    return v;
}
__device__ __forceinline__ int wave_red_i(int v) {
    #pragma unroll
    for (int m = 16; m > 0; m >>= 1) v += __shfl_xor(v, m, 32);
    return v;
}

// ---------------------------------------------------------------------------
// ws layout: [0..7] clsSum f32 | [8..15] regSum f32 | [16..23] posCnt u32
//            byte offset 256: signed char state[B*A]
// ---------------------------------------------------------------------------

__global__ void init_ws_kernel(float* __restrict__ ws) {
    int t = threadIdx.x;
    if (t < 24) ws[t] = 0.0f;   // zeroes cls/reg sums and the u32 counters
}

// Stage 1: per-anchor assignment + regression smooth-L1 for positives.
__global__ __launch_bounds__(256) void assign_kernel(
    const float* __restrict__ anchors,      // [A,4]
    const float* __restrict__ regs,         // [B,A,4]
    const float* __restrict__ anns,         // [B,M,9]
    signed char* __restrict__ state,        // [B,A]
    float* __restrict__ regSum,             // [B]
    unsigned int* __restrict__ posCnt)      // [B]
{
    __shared__ float sm[MM * 9];
    const int b   = blockIdx.y;
    const int tid = threadIdx.x;
    const float* annb = anns + (size_t)b * MM * 9;

#if ASYNC_LDS
    for (int t = tid; t < MM * 9; t += blockDim.x) {
        __builtin_amdgcn_global_load_async_to_lds_b32(
            (__attribute__((address_space(1))) int*)(annb + t),
            (__attribute__((address_space(3))) int*)(sm + t), 0, 0);
    }
    wait_async0();
#else
    for (int t = tid; t < MM * 9; t += blockDim.x) sm[t] = annb[t];
#endif
    __syncthreads();

    const int a = blockIdx.x * blockDim.x + tid;
    float r_acc = 0.0f;
    int   p_acc = 0;

    if (a < AA) {
        f4 an = ((const f4*)anchors)[a];
        const float aw = an.z - an.x, ah = an.w - an.y;
        const float areaA = aw * ah;

        float best = -1.0f; int bj = 0;
        #pragma unroll
        for (int j = 0; j < MM; ++j) {
            const float* g = sm + j * 9;
            float bx0 = g[0], by0 = g[1], bx1 = g[2], by1 = g[3];
            float iw = fmaxf(fminf(an.z, bx1) - fmaxf(an.x, bx0), 0.0f);
            float ih = fmaxf(fminf(an.w, by1) - fmaxf(an.y, by0), 0.0f);
            float inter = iw * ih;
            float ua = fmaxf(areaA + (bx1 - bx0) * (by1 - by0) - inter, 1e-8f);
            float iou = inter / ua;
            if (iou > best) { best = iou; bj = j; }   // strict > == first argmax
        }

        signed char st;
        if (best >= 0.5f) {
            const float* g = sm + bj * 9;
            st = (signed char)(1 + (int)g[4]);
            p_acc = 1;
            // regression target deltas
            float acx = an.x + 0.5f * aw, acy = an.y + 0.5f * ah;
            float gwr = g[2] - g[0], ghr = g[3] - g[1];
            float gcx = g[0] + 0.5f * gwr, gcy = g[1] + 0.5f * ghr;
            float gw = fmaxf(gwr, 1.0f), gh = fmaxf(ghr, 1.0f);
            f4 rg = ((const f4*)regs)[(size_t)b * AA + a];
            float t0 = ((gcx - acx) / aw) * 10.0f;   // /0.1
            float t1 = ((gcy - acy) / ah) * 10.0f;
            float t2 = __logf(gw / aw) * 5.0f;       // /0.2
            float t3 = __logf(gh / ah) * 5.0f;
            const float TH = 1.0f / 9.0f, HB = 0.5f / 9.0f;
            float d0 = fabsf(t0 - rg.x), d1 = fabsf(t1 - rg.y);
            float d2 = fabsf(t2 - rg.z), d3 = fabsf(t3 - rg.w);
            r_acc  = (d0 <= TH ? 4.5f * d0 * d0 : d0 - HB);
            r_acc += (d1 <= TH ? 4.5f * d1 * d1 : d1 - HB);
            r_acc += (d2 <= TH ? 4.5f * d2 * d2 : d2 - HB);
            r_acc += (d3 <= TH ? 4.5f * d3 * d3 : d3 - HB);
        } else {
            st = (best < 0.4f) ? (signed char)0 : (signed char)(-1);
        }
        state[(size_t)b * AA + a] = st;
    }

    // wave -> block reduction, one atomic pair per block
    r_acc = wave_red_f(r_acc);
    p_acc = wave_red_i(p_acc);
    __shared__ float wred[8];
    __shared__ int   wcnt[8];
    const int wid = tid >> 5, lane = tid & 31;
    if (lane == 0) { wred[wid] = r_acc; wcnt[wid] = p_acc; }
    __syncthreads();
    if (wid == 0) {
        const int nw = blockDim.x >> 5;
        float rv = (lane < nw) ? wred[lane] : 0.0f;
        int   cv = (lane < nw) ? wcnt[lane] : 0;
        rv = wave_red_f(rv);
        cv = wave_red_i(cv);
        if (lane == 0) {
            atomicAdd(&regSum[b], rv);
            atomicAdd(&posCnt[b], (unsigned int)cv);
        }
    }
}

// Stage 2: stream 256MB of classifications (NT, 128-bit), focal loss sum per image.
__global__ __launch_bounds__(256) void clsloss_kernel(
    const float* __restrict__ cls,          // [B,A,C]
    const signed char* __restrict__ state,  // [B,A]
    float* __restrict__ clsSum)             // [B]
{
    const int b = blockIdx.y;
    const int G = AA * (CC / 4);            // float4 groups per image
    const f4* base = (const f4*)(cls + (size_t)b * AA * CC);
    const signed char* stb = state + (size_t)b * AA;

    float acc = 0.0f;
    for (int idx = blockIdx.x * blockDim.x + threadIdx.x; idx < G;
         idx += gridDim.x * blockDim.x) {
        const int a = idx / (CC / 4);
        const int g = idx - a * (CC / 4);
        const int s = stb[a];
        f4 p4 = __builtin_nontemporal_load(base + idx);  // th:NT single-use stream
        if (s >= 0) {
            const int c0 = g * 4;
            float pv[4] = {p4.x, p4.y, p4.z, p4.w};
            #pragma unroll
            for (int k = 0; k < 4; ++k) {
                float p = fminf(fmaxf(pv[k], 1e-4f), 1.0f - 1e-4f);
                bool tp = (s == 1 + c0 + k);             // one-hot positive entry
                float x  = tp ? p : 1.0f - p;
                float q  = 1.0f - x;
                float al = tp ? 0.25f : 0.75f;
                acc += al * q * q * (-__logf(x));        // one log per element
            }
        }
    }

    acc = wave_red_f(acc);
    __shared__ float wred[8];
    const int wid = threadIdx.x >> 5, lane = threadIdx.x & 31;
    if (lane == 0) wred[wid] = acc;
    __syncthreads();
    if (wid == 0) {
        const int nw = blockDim.x >> 5;
        float v = (lane < nw) ? wred[lane] : 0.0f;
        v = wave_red_f(v);
        if (lane == 0) atomicAdd(&clsSum[b], v);
    }
}

// Stage 3: finalize. Per-image normalization, then the 8-image means via one
// v_wmma_f32_16x16x4_f32: cls_l[0..3]/cls_l[4..7]/reg_l[0..3]/reg_l[4..7] are
// placed along K of A-matrix rows 0..3 (K maps to {vgpr, lane-half}) and
// multiplied by an all-ones B, so D[M][*] = row sums. Single 32-lane wave,
// uniform control flow up to the WMMA (EXEC all ones).
__global__ void finalize_kernel(const float* __restrict__ ws, float* __restrict__ out)
{
    const float* clsSum = ws;
    const float* regSum = ws + 8;
    const unsigned int* cnt = (const unsigned int*)(ws + 16);

    const int L = threadIdx.x;      // 0..31
    const int bsel = L & 7;         // clamped load: no divergence
    const float c  = (float)cnt[bsel];
    const float cl = clsSum[bsel] / fmaxf(c, 1.0f);
    const float rl = (c > 0.0f) ? (regSum[bsel] / (4.0f * c)) : 0.0f;

    const int m = L & 15, h = L >> 4;
    const int b0 = (m & 1) * 4 + 2 * h;            // {0,2,4,6}
    float cv0 = __shfl(cl, b0, 32),  cv1 = __shfl(cl, b0 + 1, 32);
    float rv0 = __shfl(rl, b0, 32),  rv1 = __shfl(rl, b0 + 1, 32);
    const bool isC = (m == 0) | (m == 1);
    const bool isR = (m == 2) | (m == 3);

    v2f Aop;
    Aop.x = isC ? cv0 : (isR ? rv0 : 0.0f);
    Aop.y = isC ? cv1 : (isR ? rv1 : 0.0f);
    v2f Bop = {1.0f, 1.0f};
    v8f Cc  = {};

    v8f D = __builtin_amdgcn_wmma_f32_16x16x4_f32(
        false, Aop, false, Bop, (short)0, Cc, false, false);

    if (L == 0) {
        // lane0 holds (M=0..7, N=0): D[0]=Σcls[0..3], D[1]=Σcls[4..7],
        //                            D[2]=Σreg[0..3], D[3]=Σreg[4..7]
        out[0] = (D[0] + D[1]) * 0.125f;
        out[1] = (D[2] + D[3]) * 0.125f;
    }
}

extern "C" void kernel_launch(void* const* d_in, const int* in_sizes, int n_in,
                              void* d_out, int out_size, void* d_ws, size_t ws_size,
                              hipStream_t stream)
{
    const float* cls  = (const float*)d_in[0];   // [B,A,C]
    const float* regs = (const float*)d_in[1];   // [B,A,4]
    const float* anc  = (const float*)d_in[2];   // [1,A,4]
    const float* ann  = (const float*)d_in[3];   // [B,M,9]
    float* ws = (float*)d_ws;
    signed char* state = (signed char*)((char*)d_ws + 256);
    float* out = (float*)d_out;

    hipLaunchKernelGGL(init_ws_kernel, dim3(1), dim3(64), 0, stream, ws);

    dim3 gA((AA + 255) / 256, BB);
    hipLaunchKernelGGL(assign_kernel, gA, dim3(256), 0, stream,
                       anc, regs, ann, state, ws + 8, (unsigned int*)(ws + 16));

    dim3 gB(1024, BB);   // grid-strided: ~8 f4 loads/thread, 8192 atomics total
    hipLaunchKernelGGL(clsloss_kernel, gB, dim3(256), 0, stream, cls, state, ws);

    hipLaunchKernelGGL(finalize_kernel, dim3(1), dim3(32), 0, stream, ws, out);
}